// CommAwareGCN_8358006358160
// MI455X (gfx1250) — compile-verified
//
#include <hip/hip_runtime.h>

typedef float v2f __attribute__((ext_vector_type(2)));
typedef float v8f __attribute__((ext_vector_type(8)));

#define D_HID 128

// Y[M,N] = relu?(X[M,K] @ W[N,K]^T + bias[N])
// One wave per 16x16 output tile, fp32 WMMA, K stepped by 4.
// Ghost rows/cols (ragged M or N) use clamped indices: they compute duplicate
// values that are simply never stored, so EXEC stays all-1s through the WMMA
// and the inner loop is pure load->wmma.
__global__ void wmma_linear_kernel(const float* __restrict__ X,
                                   const float* __restrict__ W,
                                   const float* __restrict__ bias,
                                   float* __restrict__ Y,
                                   int M, int K, int N, int do_relu)
{
    const int lane = threadIdx.x & 31;
    const int wave = threadIdx.x >> 5;          // 4 waves / block
    const int ntiles = (N + 15) >> 4;
    const int nt = blockIdx.y * 4 + wave;
    if (nt >= ntiles) return;                   // uniform per wave -> safe

    const int mt   = blockIdx.x;
    const int half = lane >> 4;                 // 0: lanes 0-15, 1: lanes 16-31
    const int l    = lane & 15;

    const int m  = mt * 16 + l;                 // A row for this lane
    const int mc = (m < M) ? m : (M - 1);
    const int n  = nt * 16 + l;                 // output col for this lane
    const int nc = (n < N) ? n : (N - 1);

    const float* xrow = X + (size_t)mc * K + 2 * half;  // K = {0,1} or {2,3} slice
    const float* wrow = W + (size_t)nc * K + 2 * half;

    v8f acc = {0.f, 0.f, 0.f, 0.f, 0.f, 0.f, 0.f, 0.f};
    #pragma unroll 8
    for (int k = 0; k < K; k += 4) {
        v2f a = *(const v2f*)(xrow + k);
        v2f b = *(const v2f*)(wrow + k);
        // (neg_a, A, neg_b, B, c_mod, C, reuse_a, reuse_b)
        acc = __builtin_amdgcn_wmma_f32_16x16x4_f32(
                  false, a, false, b, (short)0, acc, false, false);
    }

    const float bv = bias[nc];
    #pragma unroll
    for (int r = 0; r < 8; ++r) {
        float v = acc[r] + bv;
        if (do_relu) v = (v > 0.f) ? v : 0.f;
        const int row = mt * 16 + r + 8 * half; // C/D layout: VGPR r -> M=r / r+8
        if (n < N && row < M)
            Y[(size_t)row * N + n] = v;
    }
}

__global__ void zero_kernel(float* __restrict__ p, long long n)
{
    long long i = (long long)blockIdx.x * blockDim.x + threadIdx.x;
    if (i < n) p[i] = 0.0f;
}

// One wave per edge: out[dst[e]][:] += Y[src[e]][:], 4 floats per lane.
// Y and out both fit in the 192MB L2, so the random access pattern stays on-chip.
__global__ void scatter_add_kernel(const float* __restrict__ Y,
                                   const int* __restrict__ src,
                                   const int* __restrict__ dst,
                                   float* __restrict__ out, int E)
{
    const long long t = (long long)blockIdx.x * blockDim.x + threadIdx.x;
    const int e = (int)(t >> 5);
    if (e >= E) return;
    const int q = (int)(t & 31);                // feature quad 0..31

    const int s = src[e];
    const int d = dst[e];
    const float4 v = ((const float4*)(Y + (size_t)s * D_HID))[q];
    float* o = out + (size_t)d * D_HID + q * 4;
    atomicAdd(o + 0, v.x);
    atomicAdd(o + 1, v.y);
    atomicAdd(o + 2, v.z);
    atomicAdd(o + 3, v.w);
}

extern "C" void kernel_launch(void* const* d_in, const int* in_sizes, int n_in,
                              void* d_out, int out_size, void* d_ws, size_t ws_size,
                              hipStream_t stream)
{
    // setup_inputs order:
    // 0 node_features [1,N,128] f32   1 W1 [128,128]  2 b1 [128]
    // 3 W2 [128,128]                  4 b2 [128]
    // 5 Wfc [40,128]                  6 bfc [40]
    // 7 edge_src [E] i32              8 edge_dst [E] i32
    const float* X0  = (const float*)d_in[0];
    const float* W1  = (const float*)d_in[1];
    const float* b1  = (const float*)d_in[2];
    const float* W2  = (const float*)d_in[3];
    const float* b2  = (const float*)d_in[4];
    const float* Wfc = (const float*)d_in[5];
    const float* bfc = (const float*)d_in[6];
    const int*   esrc = (const int*)d_in[7];
    const int*   edst = (const int*)d_in[8];

    const int Nn   = in_sizes[0] / D_HID;       // 50000
    const int E    = in_sizes[7];               // 600000
    const int Ncls = in_sizes[6];               // 40

    float* bufA = (float*)d_ws;                       // relu(linear) output [Nn,128]
    float* bufB = bufA + (size_t)Nn * D_HID;          // aggregated        [Nn,128]
    const long long hidElems = (long long)Nn * D_HID;

    const dim3 blkLin(128);
    const int mtiles = (Nn + 15) / 16;
    const dim3 grdHid(mtiles, (D_HID / 16 + 3) / 4);      // 8 col tiles -> y=2
    const dim3 grdFc(mtiles, ((Ncls + 15) / 16 + 3) / 4); // 3 col tiles -> y=1

    const int zeroBlocks = (int)((hidElems + 255) / 256);
    const int scatBlocks = (int)(((long long)E * 32 + 255) / 256);

    // ---- Layer 1: y1 = relu(X0 @ W1^T + b1)  (node-side GEMM: relu/gather commute,
    //      so GEMM runs on 50k nodes instead of 600k edges = 12x fewer FLOPs)
    wmma_linear_kernel<<<grdHid, blkLin, 0, stream>>>(X0, W1, b1, bufA,
                                                      Nn, D_HID, D_HID, 1);
    // x1[dst] += y1[src]
    zero_kernel<<<zeroBlocks, 256, 0, stream>>>(bufB, hidElems);
    scatter_add_kernel<<<scatBlocks, 256, 0, stream>>>(bufA, esrc, edst, bufB, E);

    // ---- Layer 2: y2 = relu(x1 @ W2^T + b2)
    wmma_linear_kernel<<<grdHid, blkLin, 0, stream>>>(bufB, W2, b2, bufA,
                                                      Nn, D_HID, D_HID, 1);
    // x2[dst] += y2[src]  (bufB no longer needed; re-zero and reuse)
    zero_kernel<<<zeroBlocks, 256, 0, stream>>>(bufB, hidElems);
    scatter_add_kernel<<<scatBlocks, 256, 0, stream>>>(bufA, esrc, edst, bufB, E);

    // ---- Final FC: out = x2 @ Wfc^T + bfc
    wmma_linear_kernel<<<grdFc, blkLin, 0, stream>>>(bufB, Wfc, bfc, (float*)d_out,
                                                     Nn, D_HID, Ncls, 0);
}